// GatedAttention_16475494547605
// MI455X (gfx1250) — compile-verified
//
#include <hip/hip_runtime.h>
#include <cstddef>

// Problem constants (match reference)
#define BB 64
#define LC 384
#define LQ 128
#define EE 512

typedef float v2f __attribute__((ext_vector_type(2)));
typedef float v8f __attribute__((ext_vector_type(8)));

__device__ __forceinline__ v8f wmma4(v2f a, v2f b, v8f c) {
  return __builtin_amdgcn_wmma_f32_16x16x4_f32(false, a, false, b, (short)0, c, false, false);
}
__device__ __forceinline__ float sigmoidf_(float v) { return 1.f / (1.f + expf(-v)); }

// ---------------------------------------------------------------------------
// FP32 WMMA (V_WMMA_F32_16X16X4_F32) tiled GEMM:  OUT[M,N] = IN[M,K] @ W[N,K]^T + bias
// One wave32 per 32x64 tile: 2 M-tiles x 4 N-tiles, 8 v8f accumulators.
// Software-pipelined: next K-step's 6 b64 loads issued before current wmmas.
// A layout (16x4 f32): lane m=lane&15; v0/v1 hold K = koff, koff+1, koff=(lane>>4)*2.
// B layout mirrors A with n=lane&15 and B[k][n] = W[n][k] (contiguous pair in W row).
// D layout: vgpr r -> row m0 + r + 8*(lane>=16), col n0 + (lane&15).
// ---------------------------------------------------------------------------
__global__ __launch_bounds__(32)
void gemm_bias_f32(const float* __restrict__ IN, const float* __restrict__ W,
                   const float* __restrict__ bias, float* __restrict__ OUT,
                   int M, int K, int N) {
  const int lane = threadIdx.x;
  const int m0 = blockIdx.y * 32;
  const int n0 = blockIdx.x * 64;
  const int mloc = lane & 15;
  const int koff = (lane >> 4) << 1;

  const float* aR0 = IN + (size_t)(m0 +  0 + mloc) * K + koff;
  const float* aR1 = IN + (size_t)(m0 + 16 + mloc) * K + koff;
  const float* bR0 = W + (size_t)(n0 +  0 + mloc) * K + koff;
  const float* bR1 = W + (size_t)(n0 + 16 + mloc) * K + koff;
  const float* bR2 = W + (size_t)(n0 + 32 + mloc) * K + koff;
  const float* bR3 = W + (size_t)(n0 + 48 + mloc) * K + koff;

  v8f a00 = {}, a01 = {}, a02 = {}, a03 = {};
  v8f a10 = {}, a11 = {}, a12 = {}, a13 = {};

  v2f ca0 = *(const v2f*)(aR0), ca1 = *(const v2f*)(aR1);
  v2f cb0 = *(const v2f*)(bR0), cb1 = *(const v2f*)(bR1);
  v2f cb2 = *(const v2f*)(bR2), cb3 = *(const v2f*)(bR3);

#pragma unroll 2
  for (int k = 4; k < K; k += 4) {
    v2f na0 = *(const v2f*)(aR0 + k), na1 = *(const v2f*)(aR1 + k);
    v2f nb0 = *(const v2f*)(bR0 + k), nb1 = *(const v2f*)(bR1 + k);
    v2f nb2 = *(const v2f*)(bR2 + k), nb3 = *(const v2f*)(bR3 + k);
    a00 = wmma4(ca0, cb0, a00); a01 = wmma4(ca0, cb1, a01);
    a02 = wmma4(ca0, cb2, a02); a03 = wmma4(ca0, cb3, a03);
    a10 = wmma4(ca1, cb0, a10); a11 = wmma4(ca1, cb1, a11);
    a12 = wmma4(ca1, cb2, a12); a13 = wmma4(ca1, cb3, a13);
    ca0 = na0; ca1 = na1; cb0 = nb0; cb1 = nb1; cb2 = nb2; cb3 = nb3;
  }
  a00 = wmma4(ca0, cb0, a00); a01 = wmma4(ca0, cb1, a01);
  a02 = wmma4(ca0, cb2, a02); a03 = wmma4(ca0, cb3, a03);
  a10 = wmma4(ca1, cb0, a10); a11 = wmma4(ca1, cb1, a11);
  a12 = wmma4(ca1, cb2, a12); a13 = wmma4(ca1, cb3, a13);

  const int rbase = (lane >> 4) << 3;
  const int cloc = lane & 15;
  const float bi0 = bias[n0 +  0 + cloc];
  const float bi1 = bias[n0 + 16 + cloc];
  const float bi2 = bias[n0 + 32 + cloc];
  const float bi3 = bias[n0 + 48 + cloc];
#pragma unroll
  for (int r = 0; r < 8; ++r) {
    const int rowl = m0 + rbase + r;
    float* ol = OUT + (size_t)rowl * N + n0 + cloc;
    float* oh = OUT + (size_t)(rowl + 16) * N + n0 + cloc;
    ol[ 0] = a00[r] + bi0; ol[16] = a01[r] + bi1;
    ol[32] = a02[r] + bi2; ol[48] = a03[r] + bi3;
    oh[ 0] = a10[r] + bi0; oh[16] = a11[r] + bi1;
    oh[32] = a12[r] + bi2; oh[48] = a13[r] + bi3;
  }
}

// ---------------------------------------------------------------------------
// Gate GEMM with fused epilogue: x = pre * sigmoid(pre @ Wg^T + bg)
// M=64, K=N=1024. Same 32x64 pipelined WMMA tiling.
// ---------------------------------------------------------------------------
__global__ __launch_bounds__(32)
void gate_x_kernel(const float* __restrict__ pre, const float* __restrict__ Wg,
                   const float* __restrict__ bg, float* __restrict__ x) {
  const int K = 2 * EE, N = 2 * EE;
  const int lane = threadIdx.x;
  const int m0 = blockIdx.y * 32;
  const int n0 = blockIdx.x * 64;
  const int mloc = lane & 15;
  const int koff = (lane >> 4) << 1;

  const float* aR0 = pre + (size_t)(m0 +  0 + mloc) * K + koff;
  const float* aR1 = pre + (size_t)(m0 + 16 + mloc) * K + koff;
  const float* bR0 = Wg + (size_t)(n0 +  0 + mloc) * K + koff;
  const float* bR1 = Wg + (size_t)(n0 + 16 + mloc) * K + koff;
  const float* bR2 = Wg + (size_t)(n0 + 32 + mloc) * K + koff;
  const float* bR3 = Wg + (size_t)(n0 + 48 + mloc) * K + koff;

  v8f a00 = {}, a01 = {}, a02 = {}, a03 = {};
  v8f a10 = {}, a11 = {}, a12 = {}, a13 = {};

  v2f ca0 = *(const v2f*)(aR0), ca1 = *(const v2f*)(aR1);
  v2f cb0 = *(const v2f*)(bR0), cb1 = *(const v2f*)(bR1);
  v2f cb2 = *(const v2f*)(bR2), cb3 = *(const v2f*)(bR3);

#pragma unroll 2
  for (int k = 4; k < K; k += 4) {
    v2f na0 = *(const v2f*)(aR0 + k), na1 = *(const v2f*)(aR1 + k);
    v2f nb0 = *(const v2f*)(bR0 + k), nb1 = *(const v2f*)(bR1 + k);
    v2f nb2 = *(const v2f*)(bR2 + k), nb3 = *(const v2f*)(bR3 + k);
    a00 = wmma4(ca0, cb0, a00); a01 = wmma4(ca0, cb1, a01);
    a02 = wmma4(ca0, cb2, a02); a03 = wmma4(ca0, cb3, a03);
    a10 = wmma4(ca1, cb0, a10); a11 = wmma4(ca1, cb1, a11);
    a12 = wmma4(ca1, cb2, a12); a13 = wmma4(ca1, cb3, a13);
    ca0 = na0; ca1 = na1; cb0 = nb0; cb1 = nb1; cb2 = nb2; cb3 = nb3;
  }
  a00 = wmma4(ca0, cb0, a00); a01 = wmma4(ca0, cb1, a01);
  a02 = wmma4(ca0, cb2, a02); a03 = wmma4(ca0, cb3, a03);
  a10 = wmma4(ca1, cb0, a10); a11 = wmma4(ca1, cb1, a11);
  a12 = wmma4(ca1, cb2, a12); a13 = wmma4(ca1, cb3, a13);

  const int rbase = (lane >> 4) << 3;
  const int cloc = lane & 15;
  const float bi0 = bg[n0 +  0 + cloc];
  const float bi1 = bg[n0 + 16 + cloc];
  const float bi2 = bg[n0 + 32 + cloc];
  const float bi3 = bg[n0 + 48 + cloc];
#pragma unroll
  for (int r = 0; r < 8; ++r) {
    const int rowl = m0 + rbase + r;
    const float* pl = pre + (size_t)rowl * N + n0 + cloc;
    const float* ph = pre + (size_t)(rowl + 16) * N + n0 + cloc;
    float* xl = x + (size_t)rowl * N + n0 + cloc;
    float* xh = x + (size_t)(rowl + 16) * N + n0 + cloc;
    xl[ 0] = pl[ 0] * sigmoidf_(a00[r] + bi0);
    xl[16] = pl[16] * sigmoidf_(a01[r] + bi1);
    xl[32] = pl[32] * sigmoidf_(a02[r] + bi2);
    xl[48] = pl[48] * sigmoidf_(a03[r] + bi3);
    xh[ 0] = ph[ 0] * sigmoidf_(a10[r] + bi0);
    xh[16] = ph[16] * sigmoidf_(a11[r] + bi1);
    xh[32] = ph[32] * sigmoidf_(a12[r] + bi2);
    xh[48] = ph[48] * sigmoidf_(a13[r] + bi3);
  }
}

// ---------------------------------------------------------------------------
// Attention step: one block per batch element.
//   scores[l] = sum_e h[e] * tanh(q_mult[b,l,e] + hm[b,e] + cm[b,t,e])
//   attn = softmax(scores); cell = attn @ q[b];  pre[b] = [ c[b,t,:] | cell ]
// ---------------------------------------------------------------------------
__global__ __launch_bounds__(256)
void attn_step(const float* __restrict__ h, const float* __restrict__ hm,
               const float* __restrict__ cm_all, const float* __restrict__ q_mult,
               const float* __restrict__ q, const float* __restrict__ c,
               float* __restrict__ pre, int t) {
  __shared__ float hvec[EE];
  __shared__ float hc[EE];
  __shared__ float sc[LQ];
  __shared__ float smax, ssum;

  const int b = blockIdx.x;
  const int tid = threadIdx.x;
  const int lane = tid & 31;
  const int w = tid >> 5;

  for (int i = tid; i < EE; i += 256) {
    hvec[i] = h[(size_t)b * EE + i];
    hc[i] = hm[(size_t)b * EE + i] + cm_all[((size_t)b * LC + t) * EE + i];
  }
  __syncthreads();

  for (int l = w; l < LQ; l += 8) {
    const float* qm = q_mult + ((size_t)b * LQ + l) * EE;
    float s = 0.f;
    for (int e = lane; e < EE; e += 32)
      s += hvec[e] * tanhf(qm[e] + hc[e]);
#pragma unroll
    for (int off = 16; off; off >>= 1) s += __shfl_xor(s, off, 32);
    if (lane == 0) sc[l] = s;
  }
  __syncthreads();

  if (w == 0) {
    float m = fmaxf(fmaxf(sc[lane], sc[lane + 32]), fmaxf(sc[lane + 64], sc[lane + 96]));
#pragma unroll
    for (int off = 16; off; off >>= 1) m = fmaxf(m, __shfl_xor(m, off, 32));
    if (lane == 0) smax = m;
  }
  __syncthreads();
  if (tid < LQ) sc[tid] = expf(sc[tid] - smax);
  __syncthreads();
  if (w == 0) {
    float s = sc[lane] + sc[lane + 32] + sc[lane + 64] + sc[lane + 96];
#pragma unroll
    for (int off = 16; off; off >>= 1) s += __shfl_xor(s, off, 32);
    if (lane == 0) ssum = s;
  }
  __syncthreads();
  const float inv = 1.f / ssum;

  for (int e = tid; e < EE; e += 256) {
    float cell = 0.f;
    const float* qb = q + (size_t)b * LQ * EE + e;
#pragma unroll 4
    for (int l = 0; l < LQ; ++l) cell += sc[l] * qb[(size_t)l * EE];
    pre[(size_t)b * (2 * EE) + EE + e] = cell * inv;
    pre[(size_t)b * (2 * EE) + e] = c[((size_t)b * LC + t) * EE + e];
  }
}

// ---------------------------------------------------------------------------
// Fused GRU step, 32x16 tile per wave (2 M-tiles x 3 gate slices = 6 accs):
//   gi = x @ Wih^T (+bih), gh = h @ Whh^T (+bhh), then GRU elementwise;
//   writes h_next and hs[:, t, :]. Double-buffered h.
// ---------------------------------------------------------------------------
__global__ __launch_bounds__(32)
void gru_step(const float* __restrict__ x, const float* __restrict__ h_prev,
              const float* __restrict__ Wih, const float* __restrict__ bih,
              const float* __restrict__ Whh, const float* __restrict__ bhh,
              float* __restrict__ h_next, float* __restrict__ out, int t) {
  const int Kx = 2 * EE;   // 1024
  const int Kh = EE;       // 512
  const int lane = threadIdx.x;
  const int m0 = blockIdx.y * 32;   // batch tile
  const int j0 = blockIdx.x * 16;   // hidden-col tile
  const int mloc = lane & 15;
  const int koff = (lane >> 4) << 1;

  // gi: A = x rows (2 tiles), B = Wih rows {j, E+j, 2E+j}
  const float* aX0 = x + (size_t)(m0 +  0 + mloc) * Kx + koff;
  const float* aX1 = x + (size_t)(m0 + 16 + mloc) * Kx + koff;
  const float* wr = Wih + (size_t)(0 * EE + j0 + mloc) * Kx + koff;
  const float* wz = Wih + (size_t)(1 * EE + j0 + mloc) * Kx + koff;
  const float* wn = Wih + (size_t)(2 * EE + j0 + mloc) * Kx + koff;

  v8f gir0 = {}, giz0 = {}, gin0 = {}, gir1 = {}, giz1 = {}, gin1 = {};
  {
    v2f ca0 = *(const v2f*)(aX0), ca1 = *(const v2f*)(aX1);
    v2f cbr = *(const v2f*)(wr), cbz = *(const v2f*)(wz), cbn = *(const v2f*)(wn);
#pragma unroll 2
    for (int k = 4; k < Kx; k += 4) {
      v2f na0 = *(const v2f*)(aX0 + k), na1 = *(const v2f*)(aX1 + k);
      v2f nbr = *(const v2f*)(wr + k), nbz = *(const v2f*)(wz + k), nbn = *(const v2f*)(wn + k);
      gir0 = wmma4(ca0, cbr, gir0); giz0 = wmma4(ca0, cbz, giz0); gin0 = wmma4(ca0, cbn, gin0);
      gir1 = wmma4(ca1, cbr, gir1); giz1 = wmma4(ca1, cbz, giz1); gin1 = wmma4(ca1, cbn, gin1);
      ca0 = na0; ca1 = na1; cbr = nbr; cbz = nbz; cbn = nbn;
    }
    gir0 = wmma4(ca0, cbr, gir0); giz0 = wmma4(ca0, cbz, giz0); gin0 = wmma4(ca0, cbn, gin0);
    gir1 = wmma4(ca1, cbr, gir1); giz1 = wmma4(ca1, cbz, giz1); gin1 = wmma4(ca1, cbn, gin1);
  }

  // gh: A = h_prev rows (2 tiles), B = Whh rows {j, E+j, 2E+j}
  const float* aH0 = h_prev + (size_t)(m0 +  0 + mloc) * Kh + koff;
  const float* aH1 = h_prev + (size_t)(m0 + 16 + mloc) * Kh + koff;
  const float* vr = Whh + (size_t)(0 * EE + j0 + mloc) * Kh + koff;
  const float* vz = Whh + (size_t)(1 * EE + j0 + mloc) * Kh + koff;
  const float* vn = Whh + (size_t)(2 * EE + j0 + mloc) * Kh + koff;

  v8f ghr0 = {}, ghz0 = {}, ghn0 = {}, ghr1 = {}, ghz1 = {}, ghn1 = {};
  {
    v2f ca0 = *(const v2f*)(aH0), ca1 = *(const v2f*)(aH1);
    v2f cbr = *(const v2f*)(vr), cbz = *(const v2f*)(vz), cbn = *(const v2f*)(vn);
#pragma unroll 2
    for (int k = 4; k < Kh; k += 4) {
      v2f na0 = *(const v2f*)(aH0 + k), na1 = *(const v2f*)(aH1 + k);
      v2f nbr = *(const v2f*)(vr + k), nbz = *(const v2f*)(vz + k), nbn = *(const v2f*)(vn + k);
      ghr0 = wmma4(ca0, cbr, ghr0); ghz0 = wmma4(ca0, cbz, ghz0); ghn0 = wmma4(ca0, cbn, ghn0);
      ghr1 = wmma4(ca1, cbr, ghr1); ghz1 = wmma4(ca1, cbz, ghz1); ghn1 = wmma4(ca1, cbn, ghn1);
      ca0 = na0; ca1 = na1; cbr = nbr; cbz = nbz; cbn = nbn;
    }
    ghr0 = wmma4(ca0, cbr, ghr0); ghz0 = wmma4(ca0, cbz, ghz0); ghn0 = wmma4(ca0, cbn, ghn0);
    ghr1 = wmma4(ca1, cbr, ghr1); ghz1 = wmma4(ca1, cbz, ghz1); ghn1 = wmma4(ca1, cbn, ghn1);
  }

  const int rbase = (lane >> 4) << 3;
  const int cloc = lane & 15;
  const int j = j0 + cloc;
  const float bir = bih[j], biz = bih[EE + j], bin = bih[2 * EE + j];
  const float bhr = bhh[j], bhz = bhh[EE + j], bhn = bhh[2 * EE + j];
#pragma unroll
  for (int r = 0; r < 8; ++r) {
    {
      const int row = m0 + rbase + r;
      const float rr = sigmoidf_((gir0[r] + bir) + (ghr0[r] + bhr));
      const float zz = sigmoidf_((giz0[r] + biz) + (ghz0[r] + bhz));
      const float nn = tanhf((gin0[r] + bin) + rr * (ghn0[r] + bhn));
      const float hold = h_prev[(size_t)row * EE + j];
      const float hnew = (1.f - zz) * nn + zz * hold;
      h_next[(size_t)row * EE + j] = hnew;
      out[((size_t)row * LC + t) * EE + j] = hnew;
    }
    {
      const int row = m0 + 16 + rbase + r;
      const float rr = sigmoidf_((gir1[r] + bir) + (ghr1[r] + bhr));
      const float zz = sigmoidf_((giz1[r] + biz) + (ghz1[r] + bhz));
      const float nn = tanhf((gin1[r] + bin) + rr * (ghn1[r] + bhn));
      const float hold = h_prev[(size_t)row * EE + j];
      const float hnew = (1.f - zz) * nn + zz * hold;
      h_next[(size_t)row * EE + j] = hnew;
      out[((size_t)row * LC + t) * EE + j] = hnew;
    }
  }
}

__global__ void zero_f32(float* __restrict__ p, int n) {
  int i = blockIdx.x * 256 + threadIdx.x;
  if (i < n) p[i] = 0.f;
}

// ---------------------------------------------------------------------------
// Launch: hoisted GEMMs (q_mult, cm_all) then 384 serialized steps of
// 4 stream-ordered kernels each. Graph-capture safe.
// ---------------------------------------------------------------------------
extern "C" void kernel_launch(void* const* d_in, const int* in_sizes, int n_in,
                              void* d_out, int out_size, void* d_ws, size_t ws_size,
                              hipStream_t stream) {
  const float* c   = (const float*)d_in[0];
  const float* q   = (const float*)d_in[1];
  const float* Wuq = (const float*)d_in[2];
  const float* buq = (const float*)d_in[3];
  const float* Wuc = (const float*)d_in[4];
  const float* buc = (const float*)d_in[5];
  const float* Wvq = (const float*)d_in[6];
  const float* bvq = (const float*)d_in[7];
  const float* Wg  = (const float*)d_in[8];
  const float* bg  = (const float*)d_in[9];
  const float* Wih = (const float*)d_in[10];
  const float* bih = (const float*)d_in[11];
  const float* Whh = (const float*)d_in[12];
  const float* bhh = (const float*)d_in[13];
  float* out = (float*)d_out;

  float* ws = (float*)d_ws;
  float* q_mult = ws;                                  // B*LQ*E
  float* cm_all = q_mult + (size_t)BB * LQ * EE;       // B*LC*E
  float* hA     = cm_all + (size_t)BB * LC * EE;       // B*E
  float* hB     = hA + (size_t)BB * EE;                // B*E
  float* hm     = hB + (size_t)BB * EE;                // B*E
  float* pre    = hm + (size_t)BB * EE;                // B*2E
  float* x      = pre + (size_t)BB * 2 * EE;           // B*2E

  const dim3 wave(32);

  // Hoisted projections (large WMMA GEMMs, HBM/L2-streaming)
  gemm_bias_f32<<<dim3(EE / 64, (BB * LQ) / 32), wave, 0, stream>>>(
      q, Wuq, buq, q_mult, BB * LQ, EE, EE);
  gemm_bias_f32<<<dim3(EE / 64, (BB * LC) / 32), wave, 0, stream>>>(
      c, Wuc, buc, cm_all, BB * LC, EE, EE);
  zero_f32<<<(BB * EE) / 256, 256, 0, stream>>>(hA, BB * EE);

  for (int t = 0; t < LC; ++t) {
    float* hcur = (t & 1) ? hB : hA;
    float* hnxt = (t & 1) ? hA : hB;
    gemm_bias_f32<<<dim3(EE / 64, BB / 32), wave, 0, stream>>>(
        hcur, Wvq, bvq, hm, BB, EE, EE);
    attn_step<<<BB, 256, 0, stream>>>(hcur, hm, cm_all, q_mult, q, c, pre, t);
    gate_x_kernel<<<dim3((2 * EE) / 64, BB / 32), wave, 0, stream>>>(pre, Wg, bg, x);
    gru_step<<<dim3(EE / 16, BB / 32), wave, 0, stream>>>(
        x, hcur, Wih, bih, Whh, bhh, hnxt, out, t);
  }
}